// PFMOnlyLearnable_24575802867930
// MI455X (gfx1250) — compile-verified
//
#include <hip/hip_runtime.h>
#include <cstdint>

// Problem constants (fixed by the reference/setup_inputs).
#define A_TOTAL   4096
#define N_NBR     128
#define PRED_STEPS 12
#define BLOCK     32            // one wave32 == one workgroup (barrier-free)
#define ROW_FLOATS 260          // 128 neighbors * 2 floats + 4 pad -> 1040B rows (16B aligned)

__global__ __launch_bounds__(BLOCK)
void pfm_rollout_kernel(const float* __restrict__ history,
                        const float* __restrict__ neighbors,
                        const float* __restrict__ goal,
                        float* __restrict__ out)
{
    __shared__ float smem[BLOCK * ROW_FLOATS];   // 33,280 B

    const int tid        = threadIdx.x;
    const int blockStart = blockIdx.x * BLOCK;
    const int agent      = blockStart + tid;
    const int b          = agent >> 12;          // agent / 4096

    // ---------------- Async stage: 32 rows x 1024B (global) -> 1040B rows (LDS) ----
    // 2048 16-byte chunks, lanes take consecutive chunks -> 512B coalesced per issue.
    const float*  gbase    = neighbors + (size_t)blockStart * (2 * N_NBR);
    const uint32_t lds_base = (uint32_t)(size_t)(&smem[0]);   // low 32b of LDS flat addr = LDS offset
    #pragma unroll 4
    for (int m = 0; m < 64; ++m) {
        int c = m * BLOCK + tid;
        uint32_t row  = (uint32_t)(c >> 6);
        uint32_t col  = (uint32_t)(c & 63);
        uint32_t goff = row * 1024u + col * 16u;
        uint32_t loff = lds_base + row * (ROW_FLOATS * 4u) + col * 16u;
        asm volatile("global_load_async_to_lds_b128 %0, %1, %2"
                     :: "v"(loff), "v"(goff), "s"(gbase) : "memory");
    }

    // Useful latency-hiding work while the async DMA is in flight:
    // history[agent, 6..7, :] as one aligned float4 = {x6,y6,x7,y7}
    float4 h67 = ((const float4*)history)[agent * 4 + 3];
    float cx = h67.z, cy = h67.w;                 // cur0
    float prx = cx + (h67.z - h67.x);             // pred0 = cur0 + vel
    float pry = cy + (h67.w - h67.y);
    const float gx = goal[2 * b + 0];
    const float gy = goal[2 * b + 1];

    // Wave's own async loads complete; single-wave workgroup -> no barrier needed.
    asm volatile("s_wait_asynccnt 0" ::: "memory");

    const float2* __restrict__ rowp = (const float2*)(smem + tid * ROW_FLOATS);

    float px = cx, py = cy;     // raw rollout position
    float ccx = cx, ccy = cy;   // speed-clamped trajectory
    float2* __restrict__ o = (float2*)(out + (size_t)agent * (2 * PRED_STEPS));

    const float MIN_SPEED = 0.027f * 0.85f;
    const float MAX_SPEED = 0.027f * 1.15f;
    const float DT        = 0.1f;

    for (int t = 0; t < PRED_STEPS; ++t) {
        // ---- forces at (px,py); prediction term is pred0 on step 0, then cur ----
        float fx = (gx - px) + 0.5f * (prx - px);
        float fy = (gy - py) + 0.5f * (pry - py);
        float rx = 0.0f, ry = 0.0f;
        #pragma unroll 4
        for (int k = 0; k < N_NBR; ++k) {
            float2 nb = rowp[k];
            float dx = px - nb.x;
            float dy = py - nb.y;
            float d2 = dx * dx + dy * dy;
            float d  = __builtin_amdgcn_sqrtf(d2) + 1e-6f;     // v_sqrt_f32
            float w  = __builtin_amdgcn_rcpf(d * d);           // v_rcp_f32
            w = (d < 0.5f) ? w : 0.0f;                         // repulsion mask, branchless
            rx = fmaf(dx, w, rx);
            ry = fmaf(dy, w, ry);
        }
        fx += rx;
        fy += ry;
        px = fmaf(fx, DT, px);
        py = fmaf(fy, DT, py);
        prx = px; pry = py;     // subsequent steps: pred == cur

        // ---- fused speed-clamp scan against the running clamped position ----
        float dxs = px - ccx, dys = py - ccy;
        float sp  = __builtin_amdgcn_sqrtf(dxs * dxs + dys * dys);
        float spf = (sp < MIN_SPEED) ? MIN_SPEED : ((sp > MAX_SPEED) ? MAX_SPEED : sp);
        spf = (sp > 0.0f) ? spf : sp;
        float scale = spf * __builtin_amdgcn_rcpf(sp + 1e-8f);
        ccx = fmaf(dxs, scale, ccx);
        ccy = fmaf(dys, scale, ccy);
        o[t] = make_float2(ccx, ccy);
    }
}

extern "C" void kernel_launch(void* const* d_in, const int* in_sizes, int n_in,
                              void* d_out, int out_size, void* d_ws, size_t ws_size,
                              hipStream_t stream) {
    const float* history   = (const float*)d_in[0];   // (8, 4096, 8, 2)
    const float* neighbors = (const float*)d_in[1];   // (8, 4096, 128, 2)
    const float* goal      = (const float*)d_in[2];   // (8, 2)
    float* out             = (float*)d_out;           // (8, 4096, 12, 2)

    const int agents = in_sizes[0] / (8 * 2);         // B*A = 32768
    dim3 grid(agents / BLOCK);
    pfm_rollout_kernel<<<grid, BLOCK, 0, stream>>>(history, neighbors, goal, out);
}